// GATWithEdgeFeatures_42949672960528
// MI455X (gfx1250) — compile-verified
//
#include <hip/hip_runtime.h>

#define N_NODES_C   20000
#define N_EDGES_C   320000
#define EDGE_F_C    16
#define OUT_F_C     64
#define NUM_CLASSES_C 10
#define NUM_GRAPHS_C 40
#define ALPHA_C     0.2f

typedef float v2f __attribute__((ext_vector_type(2)));
typedef float v8f __attribute__((ext_vector_type(8)));

// ------------------------------------------------------------------
// Row-major fp32 GEMM  C[M,N] = A[M,K] @ B[K,N]  via v_wmma_f32_16x16x4_f32.
// Each wave owns FIVE 16x16 C tiles stacked in M (80 rows) sharing one
// B fragment per K-step: 5 independent WMMAs per 6 fragment loads, and the
// independent accumulators avoid the WMMA->WMMA RAW hazard chain.
// Requires M % 80 == 0 (20000 = 250*80), N % 16 == 0, K % 4 == 0.
// Wave-uniform early exit keeps EXEC all-ones at every WMMA.
// ------------------------------------------------------------------
__global__ __launch_bounds__(256) void gemm_wmma(const float* __restrict__ A,
                                                 const float* __restrict__ B,
                                                 float* __restrict__ C,
                                                 int M, int N, int K)
{
    const int lane = threadIdx.x & 31;
    const int wave = threadIdx.x >> 5;
    const int grp  = blockIdx.y * 8 + wave;      // group of 5 M-tiles
    if (grp * 80 >= M) return;                   // wave-uniform
    const int  n0   = blockIdx.x * 16;
    const int  hi   = (lane >= 16) ? 1 : 0;
    const int  lr   = lane & 15;
    const int  koff = hi * 2;                    // lanes 0-15: K+0/K+1, 16-31: K+2/K+3
    const int  ncol = n0 + lr;
    const long mbase = (long)grp * 80 + lr;

    const float* Ap0 = A + (mbase +  0) * K + koff;
    const float* Ap1 = A + (mbase + 16) * K + koff;
    const float* Ap2 = A + (mbase + 32) * K + koff;
    const float* Ap3 = A + (mbase + 48) * K + koff;
    const float* Ap4 = A + (mbase + 64) * K + koff;

    v8f acc0 = {}, acc1 = {}, acc2 = {}, acc3 = {}, acc4 = {};
    for (int k = 0; k < K; k += 4) {
        v2f b;
        b.x = B[(long)(k + koff)     * N + ncol];
        b.y = B[(long)(k + koff + 1) * N + ncol];
        v2f a0; a0.x = Ap0[k]; a0.y = Ap0[k + 1];
        v2f a1; a1.x = Ap1[k]; a1.y = Ap1[k + 1];
        v2f a2; a2.x = Ap2[k]; a2.y = Ap2[k + 1];
        v2f a3; a3.x = Ap3[k]; a3.y = Ap3[k + 1];
        v2f a4; a4.x = Ap4[k]; a4.y = Ap4[k + 1];
        acc0 = __builtin_amdgcn_wmma_f32_16x16x4_f32(false, a0, false, b, (short)0, acc0, false, false);
        acc1 = __builtin_amdgcn_wmma_f32_16x16x4_f32(false, a1, false, b, (short)0, acc1, false, false);
        acc2 = __builtin_amdgcn_wmma_f32_16x16x4_f32(false, a2, false, b, (short)0, acc2, false, false);
        acc3 = __builtin_amdgcn_wmma_f32_16x16x4_f32(false, a3, false, b, (short)0, acc3, false, false);
        acc4 = __builtin_amdgcn_wmma_f32_16x16x4_f32(false, a4, false, b, (short)0, acc4, false, false);
    }
    // C layout: VGPR r -> rows r (lanes 0-15) and r+8 (lanes 16-31)
    const long crow = (long)grp * 80 + hi * 8;
#pragma unroll
    for (int r = 0; r < 8; ++r) C[(crow +  0 + r) * N + ncol] = acc0[r];
#pragma unroll
    for (int r = 0; r < 8; ++r) C[(crow + 16 + r) * N + ncol] = acc1[r];
#pragma unroll
    for (int r = 0; r < 8; ++r) C[(crow + 32 + r) * N + ncol] = acc2[r];
#pragma unroll
    for (int r = 0; r < 8; ++r) C[(crow + 48 + r) * N + ncol] = acc3[r];
#pragma unroll
    for (int r = 0; r < 8; ++r) C[(crow + 64 + r) * N + ncol] = acc4[r];
}

// ------------------------------------------------------------------
// s_i[n,h] = <Wh[n,h,:], a[h, 0:F]> ;  s_j[n,h] = <Wh[n,h,:], a[h, F:2F]>
// ------------------------------------------------------------------
template <int H, int F>
__global__ void node_scores(const float* __restrict__ Wh, const float* __restrict__ a,
                            float* __restrict__ si, float* __restrict__ sj, int N)
{
    const int t = blockIdx.x * blockDim.x + threadIdx.x;
    if (t >= N * H) return;
    const int n = t / H, h = t % H;
    const float* w  = Wh + (long)n * H * F + (long)h * F;
    const float* ai = a + (long)h * (2 * F + EDGE_F_C);
    const float* aj = ai + F;
    float x = 0.f, y = 0.f;
#pragma unroll 4
    for (int d = 0; d < F; ++d) { x += w[d] * ai[d]; y += w[d] * aj[d]; }
    si[t] = x;
    sj[t] = y;
}

// ------------------------------------------------------------------
// Thread-per-edge: e = LeakyReLU(s_i[src]+s_j[dst]+edge_attr@a_e^T),
// softmax over the H heads (reference does softmax on axis=1), store att.
// ------------------------------------------------------------------
template <int H, int F>
__global__ void edge_att_kernel(const float* __restrict__ si, const float* __restrict__ sj,
                                const float* __restrict__ edge_attr, const float* __restrict__ a,
                                const int* __restrict__ src, const int* __restrict__ dst,
                                float* __restrict__ att, int nEdges)
{
    const int e = blockIdx.x * blockDim.x + threadIdx.x;
    if (e >= nEdges) return;
    const int s = src[e], d = dst[e];

    float ea[EDGE_F_C];
    const float4* eap = (const float4*)(edge_attr + (long)e * EDGE_F_C);
#pragma unroll
    for (int q = 0; q < EDGE_F_C / 4; ++q) {
        const float4 v = eap[q];
        ea[4 * q + 0] = v.x; ea[4 * q + 1] = v.y;
        ea[4 * q + 2] = v.z; ea[4 * q + 3] = v.w;
    }
    float sc[H];
#pragma unroll
    for (int h = 0; h < H; ++h) {
        const float* ae = a + (long)h * (2 * F + EDGE_F_C) + 2 * F;
        float se = 0.f;
#pragma unroll
        for (int k = 0; k < EDGE_F_C; ++k) se += ea[k] * ae[k];
        const float v = si[(long)s * H + h] + sj[(long)d * H + h] + se;
        sc[h] = (v > 0.f) ? v : ALPHA_C * v;
    }
    float mx = sc[0];
#pragma unroll
    for (int h = 1; h < H; ++h) mx = fmaxf(mx, sc[h]);
    float sum = 0.f;
#pragma unroll
    for (int h = 0; h < H; ++h) { sc[h] = __expf(sc[h] - mx); sum += sc[h]; }
    const float inv = 1.f / sum;
#pragma unroll
    for (int h = 0; h < H; ++h) att[(long)e * H + h] = sc[h] * inv;
}

// ------------------------------------------------------------------
// Wave-per-edge: out[src] += att[e] * Wh[dst]  (coalesced gather + L2 atomics)
// ------------------------------------------------------------------
template <int H, int F>
__global__ __launch_bounds__(256) void edge_apply_kernel(
    const float* __restrict__ Wh, const float* __restrict__ att,
    const int* __restrict__ src, const int* __restrict__ dst,
    float* __restrict__ out, int nEdges)
{
    const int lane = threadIdx.x & 31;
    const int wave = threadIdx.x >> 5;
    const int e    = blockIdx.x * 8 + wave;
    if (e >= nEdges) return;
    const int s = src[e], d = dst[e];
    const float* w = Wh + (long)d * H * F;
    float*       o = out + (long)s * H * F;
    // cover the H*F-float row with cacheline prefetches before the att loads
    __builtin_prefetch(&w[(long)lane * (H * F) / 32], 0, 3);
#pragma unroll
    for (int h = 0; h < H; ++h) {
        const float ah = att[(long)e * H + h];
#pragma unroll
        for (int dd = lane; dd < F; dd += 32)
            atomicAdd(&o[h * F + dd], ah * w[h * F + dd]);
    }
}

__global__ void elu_kernel(float* __restrict__ p, int n)
{
    const int i = blockIdx.x * blockDim.x + threadIdx.x;
    if (i >= n) return;
    const float v = p[i];
    p[i] = (v > 0.f) ? v : (__expf(v) - 1.f);
}

__global__ void zero_kernel(float* __restrict__ p, int n)
{
    const int i = blockIdx.x * blockDim.x + threadIdx.x;
    if (i < n) p[i] = 0.f;
}

__global__ void pool_sum_kernel(const float* __restrict__ h3, const int* __restrict__ batch,
                                float* __restrict__ sums, int N)
{
    const int i = blockIdx.x * blockDim.x + threadIdx.x;
    if (i >= N * OUT_F_C) return;
    const int n = i / OUT_F_C, d = i % OUT_F_C;
    atomicAdd(&sums[(long)batch[n] * OUT_F_C + d], h3[i]);
}

__global__ void pool_cnt_kernel(const int* __restrict__ batch, float* __restrict__ cnts, int N)
{
    const int n = blockIdx.x * blockDim.x + threadIdx.x;
    if (n >= N) return;
    atomicAdd(&cnts[batch[n]], 1.0f);
}

__global__ void classify_kernel(const float* __restrict__ sums, const float* __restrict__ cnts,
                                const float* __restrict__ W, const float* __restrict__ b,
                                float* __restrict__ out)
{
    const int g = threadIdx.x;
    if (g >= NUM_GRAPHS_C) return;
    const float inv = 1.0f / fmaxf(cnts[g], 1.0f);
    float logits[NUM_CLASSES_C];
#pragma unroll
    for (int c = 0; c < NUM_CLASSES_C; ++c) logits[c] = b[c];
    for (int d = 0; d < OUT_F_C; ++d) {
        const float p = sums[(long)g * OUT_F_C + d] * inv;
#pragma unroll
        for (int c = 0; c < NUM_CLASSES_C; ++c)
            logits[c] += p * W[(long)d * NUM_CLASSES_C + c];
    }
    float mx = logits[0];
#pragma unroll
    for (int c = 1; c < NUM_CLASSES_C; ++c) mx = fmaxf(mx, logits[c]);
    float s = 0.f;
#pragma unroll
    for (int c = 0; c < NUM_CLASSES_C; ++c) s += __expf(logits[c] - mx);
    const float lse = mx + __logf(s);
#pragma unroll
    for (int c = 0; c < NUM_CLASSES_C; ++c)
        out[(long)g * NUM_CLASSES_C + c] = logits[c] - lse;
}

// ------------------------------------------------------------------
extern "C" void kernel_launch(void* const* d_in, const int* in_sizes, int n_in,
                              void* d_out, int out_size, void* d_ws, size_t ws_size,
                              hipStream_t stream)
{
    const float* x          = (const float*)d_in[0];
    const int*   edge_index = (const int*)  d_in[1];
    const float* edge_attr  = (const float*)d_in[2];
    const int*   batch      = (const int*)  d_in[3];
    const float* W0    = (const float*)d_in[4];
    const float* a0    = (const float*)d_in[5];
    const float* W1    = (const float*)d_in[6];
    const float* a1    = (const float*)d_in[7];
    const float* W2    = (const float*)d_in[8];
    const float* a2    = (const float*)d_in[9];
    const float* cls_w = (const float*)d_in[10];
    const float* cls_b = (const float*)d_in[11];
    float* out = (float*)d_out;

    char* ws = (char*)d_ws;
    float* bufA = (float*)(ws);                 // 20000*256 f32 = 20.48 MB (Wh)
    float* bufB = (float*)(ws + 20480000);      // 20000*256 f32 (h_prime / next input)
    float* si   = (float*)(ws + 40960000);      // 20000*4
    float* sj   = (float*)(ws + 41280000);      // 20000*4
    float* attb = (float*)(ws + 41600000);      // 320000*4 = 5.12 MB
    float* psum = (float*)(ws + 46720000);      // 40*64
    float* pcnt = (float*)(ws + 46730240);      // 40

    const int* src = edge_index;                // row 0 of (2, E)
    const int* dst = edge_index + N_EDGES_C;    // row 1

    const dim3 blk256(256);
    const int gemmBlocksY = (N_NODES_C / 80 + 7) / 8;   // 250 groups -> 32
    const int edgeWaveBlocks = (N_EDGES_C + 7) / 8;
    const int edgeThrBlocks  = (N_EDGES_C + 255) / 256;

    // ---------------- Layer 1: x(20000x256) @ W0(256x256) ----------------
    gemm_wmma<<<dim3(256 / 16, gemmBlocksY), blk256, 0, stream>>>(x, W0, bufA,
        N_NODES_C, 256, 256);
    node_scores<4, 64><<<(N_NODES_C * 4 + 255) / 256, blk256, 0, stream>>>(
        bufA, a0, si, sj, N_NODES_C);
    edge_att_kernel<4, 64><<<edgeThrBlocks, blk256, 0, stream>>>(
        si, sj, edge_attr, a0, src, dst, attb, N_EDGES_C);
    zero_kernel<<<(N_NODES_C * 256 + 255) / 256, blk256, 0, stream>>>(bufB, N_NODES_C * 256);
    edge_apply_kernel<4, 64><<<edgeWaveBlocks, blk256, 0, stream>>>(
        bufA, attb, src, dst, bufB, N_EDGES_C);
    elu_kernel<<<(N_NODES_C * 256 + 255) / 256, blk256, 0, stream>>>(bufB, N_NODES_C * 256);

    // ---------------- Layer 2: h1 @ W1(256x256) ----------------
    gemm_wmma<<<dim3(256 / 16, gemmBlocksY), blk256, 0, stream>>>(bufB, W1, bufA,
        N_NODES_C, 256, 256);
    node_scores<4, 64><<<(N_NODES_C * 4 + 255) / 256, blk256, 0, stream>>>(
        bufA, a1, si, sj, N_NODES_C);
    edge_att_kernel<4, 64><<<edgeThrBlocks, blk256, 0, stream>>>(
        si, sj, edge_attr, a1, src, dst, attb, N_EDGES_C);
    zero_kernel<<<(N_NODES_C * 256 + 255) / 256, blk256, 0, stream>>>(bufB, N_NODES_C * 256);
    edge_apply_kernel<4, 64><<<edgeWaveBlocks, blk256, 0, stream>>>(
        bufA, attb, src, dst, bufB, N_EDGES_C);
    elu_kernel<<<(N_NODES_C * 256 + 255) / 256, blk256, 0, stream>>>(bufB, N_NODES_C * 256);

    // ---------------- Layer 3: h2 @ W2(256x64), heads=1, mean (=identity) --
    gemm_wmma<<<dim3(64 / 16, gemmBlocksY), blk256, 0, stream>>>(bufB, W2, bufA,
        N_NODES_C, 64, 256);
    node_scores<1, 64><<<(N_NODES_C + 255) / 256, blk256, 0, stream>>>(
        bufA, a2, si, sj, N_NODES_C);
    edge_att_kernel<1, 64><<<edgeThrBlocks, blk256, 0, stream>>>(
        si, sj, edge_attr, a2, src, dst, attb, N_EDGES_C);
    zero_kernel<<<(N_NODES_C * 64 + 255) / 256, blk256, 0, stream>>>(bufB, N_NODES_C * 64);
    edge_apply_kernel<1, 64><<<edgeWaveBlocks, blk256, 0, stream>>>(
        bufA, attb, src, dst, bufB, N_EDGES_C);
    // concat=False, heads=1 -> mean over heads is identity, no activation

    // ---------------- Pool + classify ----------------
    zero_kernel<<<(NUM_GRAPHS_C * OUT_F_C + 255) / 256, blk256, 0, stream>>>(
        psum, NUM_GRAPHS_C * OUT_F_C);
    zero_kernel<<<1, blk256, 0, stream>>>(pcnt, NUM_GRAPHS_C);
    pool_sum_kernel<<<(N_NODES_C * OUT_F_C + 255) / 256, blk256, 0, stream>>>(
        bufB, batch, psum, N_NODES_C);
    pool_cnt_kernel<<<(N_NODES_C + 255) / 256, blk256, 0, stream>>>(batch, pcnt, N_NODES_C);
    classify_kernel<<<1, 64, 0, stream>>>(psum, pcnt, cls_w, cls_b, out);
}